// LRU_50852412785331
// MI455X (gfx1250) — compile-verified
//
#include <hip/hip_runtime.h>
#include <math.h>

typedef __attribute__((ext_vector_type(2))) float v2f;
typedef __attribute__((ext_vector_type(8))) float v8f;

#define H_DIM 512
#define V_DIM 512
#define SEQ_L 2048
#define N_CHUNK 16
#define CHUNK_LEN (SEQ_L / N_CHUNK)

// ---------------------------------------------------------------------------
// LayerNorm: one wave (32 lanes) per row of 512 floats.
// ---------------------------------------------------------------------------
__global__ __launch_bounds__(256) void ln_kernel(const float* __restrict__ x,
                                                 float* __restrict__ xs) {
    int row  = blockIdx.x * 8 + (threadIdx.x >> 5);
    int lane = threadIdx.x & 31;
    const float* xr = x + (size_t)row * H_DIM;
    float vals[16];
    float s = 0.f, s2 = 0.f;
#pragma unroll
    for (int i = 0; i < 16; ++i) {
        float v = xr[lane + 32 * i];
        vals[i] = v; s += v; s2 += v * v;
    }
#pragma unroll
    for (int off = 16; off > 0; off >>= 1) {
        s  += __shfl_xor(s,  off, 32);
        s2 += __shfl_xor(s2, off, 32);
    }
    float mu  = s * (1.0f / H_DIM);
    float var = s2 * (1.0f / H_DIM) - mu * mu;
    float inv = rsqrtf(var + 1e-5f);
    float* orow = xs + (size_t)row * H_DIM;
#pragma unroll
    for (int i = 0; i < 16; ++i)
        orow[lane + 32 * i] = (vals[i] - mu) * inv;
}

// ---------------------------------------------------------------------------
// Generic fp32 WMMA GEMM:  C[M,N] = A[M,K] @ B[K,N]  (row-major all)
// block tile 128x64, BK=32, 8 waves, each wave -> 16 rows x 64 cols (4 accs).
// Tiles staged with CDNA5 async global->LDS copies (ASYNCcnt), double-buffered
// so tile k+1 streams in under the WMMAs of tile k.
// mode 0: C = acc
// mode 1: C = acc * expf(scale_log[col])                 (gamma epilogue)
// mode 2: C = gmat[row,col] * sigmoid(acc + bias[col])   (half-GLU gate)
// ---------------------------------------------------------------------------
#define ASTR 36   // LDS stride for A tile (16B multiple, conflict-free frag reads)
#define BSTR 80   // LDS stride for B tile

__global__ __launch_bounds__(256) void gemm_wmma_f32(
    const float* __restrict__ A, const float* __restrict__ Bm,
    float* __restrict__ C, int M, int N, int K, int mode,
    const float* __restrict__ scale_log, const float* __restrict__ gmat,
    const float* __restrict__ bias) {
    __shared__ __align__(16) float As[2][128 * ASTR];
    __shared__ __align__(16) float Bs[2][32 * BSTR];

    int tid  = threadIdx.x;
    int lane = tid & 31;
    int wave = tid >> 5;          // 0..7
    int bn   = blockIdx.x;        // N/64 tiles
    int bm   = blockIdx.y;        // M/128 tiles

    int lane15 = lane & 15;
    int half   = lane >> 4;       // 0 or 1
    int koff2  = half << 1;       // 0 or 2  (K selection within a 4-step)

    // --- per-thread async transfer descriptors (advance one K-tile per issue)
    uint64_t aAddr[4];
    uint32_t aLds[2][4];
#pragma unroll
    for (int i = 0; i < 4; ++i) {
        int idx4 = tid + 256 * i;            // 0..1023
        int r    = idx4 >> 3;                // 0..127
        int c4   = (idx4 & 7) << 2;          // 0,4,...,28
        aAddr[i]   = (uint64_t)(uintptr_t)(A + (size_t)(bm * 128 + r) * K + c4);
        aLds[0][i] = (uint32_t)(uintptr_t)&As[0][r * ASTR + c4];
        aLds[1][i] = (uint32_t)(uintptr_t)&As[1][r * ASTR + c4];
    }
    uint64_t bAddr[2];
    uint32_t bLds[2][2];
#pragma unroll
    for (int i = 0; i < 2; ++i) {
        int idx4 = tid + 256 * i;            // 0..511
        int r    = idx4 >> 4;                // 0..31
        int c4   = (idx4 & 15) << 2;         // 0..60
        bAddr[i]   = (uint64_t)(uintptr_t)(Bm + (size_t)r * N + bn * 64 + c4);
        bLds[0][i] = (uint32_t)(uintptr_t)&Bs[0][r * BSTR + c4];
        bLds[1][i] = (uint32_t)(uintptr_t)&Bs[1][r * BSTR + c4];
    }
    const uint64_t aStep = 32u * 4u;               // advance K by 32 floats
    const uint64_t bStep = (uint64_t)32u * N * 4u; // advance 32 rows of B

    v8f acc[4];
#pragma unroll
    for (int s = 0; s < 4; ++s) acc[s] = (v8f){0, 0, 0, 0, 0, 0, 0, 0};

    // prologue: stream tile 0 into buffer 0
#pragma unroll
    for (int i = 0; i < 4; ++i) {
        asm volatile("global_load_async_to_lds_b128 %0, %1, off"
                     :: "v"(aLds[0][i]), "v"(aAddr[i]) : "memory");
        aAddr[i] += aStep;
    }
#pragma unroll
    for (int i = 0; i < 2; ++i) {
        asm volatile("global_load_async_to_lds_b128 %0, %1, off"
                     :: "v"(bLds[0][i]), "v"(bAddr[i]) : "memory");
        bAddr[i] += bStep;
    }
    asm volatile("s_wait_asynccnt 0x0" ::: "memory");
    __syncthreads();

    int KT = K >> 5;
    for (int kt = 0; kt < KT; ++kt) {
        int buf = kt & 1;
        // issue async copies for tile kt+1 into the other buffer
        if (kt + 1 < KT) {
            int nb = buf ^ 1;
#pragma unroll
            for (int i = 0; i < 4; ++i) {
                asm volatile("global_load_async_to_lds_b128 %0, %1, off"
                             :: "v"(aLds[nb][i]), "v"(aAddr[i]) : "memory");
                aAddr[i] += aStep;
            }
#pragma unroll
            for (int i = 0; i < 2; ++i) {
                asm volatile("global_load_async_to_lds_b128 %0, %1, off"
                             :: "v"(bLds[nb][i]), "v"(bAddr[i]) : "memory");
                bAddr[i] += bStep;
            }
        }

        const float* Asb = &As[buf][0];
        const float* Bsb = &Bs[buf][0];
#pragma unroll
        for (int kk = 0; kk < 8; ++kk) {
            int kb = (kk << 2) + koff2;
            v2f a;
            a.x = Asb[(wave * 16 + lane15) * ASTR + kb];
            a.y = Asb[(wave * 16 + lane15) * ASTR + kb + 1];
#pragma unroll
            for (int s = 0; s < 4; ++s) {
                v2f b;
                int ncol = s * 16 + lane15;
                b.x = Bsb[kb * BSTR + ncol];
                b.y = Bsb[(kb + 1) * BSTR + ncol];
                acc[s] = __builtin_amdgcn_wmma_f32_16x16x4_f32(
                    false, a, false, b, (short)0, acc[s], false, false);
            }
        }

        // next tile's async copies must be complete (all waves) before reuse
        asm volatile("s_wait_asynccnt 0x0" ::: "memory");
        __syncthreads();
    }

    // --- epilogue / store
#pragma unroll
    for (int s = 0; s < 4; ++s) {
        int col = bn * 64 + s * 16 + lane15;
        float gam = 1.0f, bb = 0.0f;
        if (mode == 1) gam = expf(scale_log[col]);
        if (mode == 2) bb  = bias[col];
#pragma unroll
        for (int j = 0; j < 8; ++j) {
            int row = bm * 128 + wave * 16 + j + 8 * half;
            size_t o = (size_t)row * N + col;
            float v = acc[s][j];
            if (mode == 0) {
                C[o] = v;
            } else if (mode == 1) {
                C[o] = v * gam;
            } else {
                float z = v + bb;
                float sg = 1.0f / (1.0f + expf(-z));
                C[o] = gmat[o] * sg;
            }
        }
    }
}

// ---------------------------------------------------------------------------
// LRU scan, chunked.  lam = exp(-exp(nu)) * exp(i*exp(theta)).
// ---------------------------------------------------------------------------
__device__ inline void lam_of(const float* nu_log, const float* theta_log,
                              int v, float& lr, float& li) {
    float nu    = expf(nu_log[v]);
    float theta = expf(theta_log[v]);
    float mag   = expf(-nu);
    lr = mag * cosf(theta);
    li = mag * sinf(theta);
}

// phase A: per-chunk aggregate  b_c = sum_j lam^{n-1-j} u_j
__global__ __launch_bounds__(256) void scan_chunks(
    const float* __restrict__ u_re, const float* __restrict__ u_im,
    const float* __restrict__ nu_log, const float* __restrict__ theta_log,
    float* __restrict__ agg_re, float* __restrict__ agg_im) {
    int tid = blockIdx.x * 256 + threadIdx.x;    // B*N_CHUNK*V
    int v = tid & (V_DIM - 1);
    int c = (tid >> 9) & (N_CHUNK - 1);
    int b = tid >> 13;
    float lr, li; lam_of(nu_log, theta_log, v, lr, li);
    int t0 = b * SEQ_L + c * CHUNK_LEN;
    float br = 0.f, bi = 0.f;
    for (int j = 0; j < CHUNK_LEN; ++j) {
        size_t idx = (size_t)(t0 + j) * V_DIM + v;
        float nr = lr * br - li * bi + u_re[idx];
        float ni = lr * bi + li * br + u_im[idx];
        br = nr; bi = ni;
    }
    agg_re[tid] = br; agg_im[tid] = bi;
}

// phase B: carry propagation across chunks; carry_in[c] = h at end of chunk c-1
__global__ __launch_bounds__(256) void scan_carry(
    const float* __restrict__ agg_re, const float* __restrict__ agg_im,
    const float* __restrict__ nu_log, const float* __restrict__ theta_log,
    float* __restrict__ car_re, float* __restrict__ car_im) {
    int tid = blockIdx.x * 256 + threadIdx.x;    // B*V
    int v = tid & (V_DIM - 1);
    int b = tid >> 9;
    float lr, li; lam_of(nu_log, theta_log, v, lr, li);
    // lam^CHUNK_LEN (128) via 7 complex squarings
    float ar = lr, ai = li;
#pragma unroll
    for (int i = 0; i < 7; ++i) {
        float nr = ar * ar - ai * ai;
        float ni = 2.f * ar * ai;
        ar = nr; ai = ni;
    }
    float cr = 0.f, ci = 0.f;
    for (int c = 0; c < N_CHUNK; ++c) {
        int o = (b * N_CHUNK + c) * V_DIM + v;
        car_re[o] = cr; car_im[o] = ci;
        float nr = ar * cr - ai * ci + agg_re[o];
        float ni = ar * ci + ai * cr + agg_im[o];
        cr = nr; ci = ni;
    }
}

// phase C: re-run each chunk with its carry, write hcat [M, 2V] (re | im)
__global__ __launch_bounds__(256) void scan_apply(
    const float* __restrict__ u_re, const float* __restrict__ u_im,
    const float* __restrict__ nu_log, const float* __restrict__ theta_log,
    const float* __restrict__ car_re, const float* __restrict__ car_im,
    float* __restrict__ hcat) {
    int tid = blockIdx.x * 256 + threadIdx.x;
    int v = tid & (V_DIM - 1);
    int c = (tid >> 9) & (N_CHUNK - 1);
    int b = tid >> 13;
    float lr, li; lam_of(nu_log, theta_log, v, lr, li);
    int t0 = b * SEQ_L + c * CHUNK_LEN;
    float hr = car_re[tid], hi = car_im[tid];
    for (int j = 0; j < CHUNK_LEN; ++j) {
        size_t idx = (size_t)(t0 + j) * V_DIM + v;
        float nr = lr * hr - li * hi + u_re[idx];
        float ni = lr * hi + li * hr + u_im[idx];
        hr = nr; hi = ni;
        size_t orow = (size_t)(t0 + j) * (2 * V_DIM);
        hcat[orow + v]         = hr;
        hcat[orow + V_DIM + v] = hi;
    }
}

// ---------------------------------------------------------------------------
// pack B2 = [C_re ; -C_im]  (1024 x 512, row-major)
// ---------------------------------------------------------------------------
__global__ __launch_bounds__(256) void pack_b2(const float* __restrict__ C_re,
                                               const float* __restrict__ C_im,
                                               float* __restrict__ B2) {
    int tid = blockIdx.x * 256 + threadIdx.x;     // 1024*512
    int k = tid >> 9;
    int n = tid & (H_DIM - 1);
    B2[tid] = (k < V_DIM) ? C_re[k * H_DIM + n]
                          : -C_im[(k - V_DIM) * H_DIM + n];
}

// ---------------------------------------------------------------------------
// transpose W (512x512):  WT[k,n] = W[n,k]
// ---------------------------------------------------------------------------
__global__ __launch_bounds__(256) void transpose_k(const float* __restrict__ W,
                                                   float* __restrict__ WT) {
    __shared__ float tile[32][33];
    int tx = threadIdx.x & 31;
    int ty = threadIdx.x >> 5;    // 0..7
#pragma unroll
    for (int p = 0; p < 4; ++p) {
        int r = blockIdx.y * 32 + ty + p * 8;
        tile[ty + p * 8][tx] = W[(size_t)r * H_DIM + blockIdx.x * 32 + tx];
    }
    __syncthreads();
#pragma unroll
    for (int p = 0; p < 4; ++p) {
        int r = blockIdx.x * 32 + ty + p * 8;
        WT[(size_t)r * H_DIM + blockIdx.y * 32 + tx] = tile[tx][ty + p * 8];
    }
}

// ---------------------------------------------------------------------------
extern "C" void kernel_launch(void* const* d_in, const int* in_sizes, int n_in,
                              void* d_out, int out_size, void* d_ws, size_t ws_size,
                              hipStream_t stream) {
    const float* x         = (const float*)d_in[0];
    const float* nu_log    = (const float*)d_in[1];
    const float* theta_log = (const float*)d_in[2];
    const float* gamma_log = (const float*)d_in[3];
    const float* B_re      = (const float*)d_in[4];
    const float* B_im      = (const float*)d_in[5];
    const float* C_re      = (const float*)d_in[6];
    const float* C_im      = (const float*)d_in[7];
    const float* W_dense   = (const float*)d_in[8];
    const float* b_dense   = (const float*)d_in[9];
    float* out = (float*)d_out;

    const int M = in_sizes[0] / H_DIM;            // 16384 tokens
    float* ws   = (float*)d_ws;
    float* xs     = ws;                              // M*512   (reused as y)
    float* u_re   = xs   + (size_t)M * H_DIM;        // M*512
    float* u_im   = u_re + (size_t)M * V_DIM;        // M*512
    float* hcat   = u_im + (size_t)M * V_DIM;        // M*1024
    float* B2     = hcat + (size_t)M * 2 * V_DIM;    // 1024*512
    float* WT     = B2   + 2 * V_DIM * H_DIM;        // 512*512
    float* agg_re = WT   + H_DIM * H_DIM;            // 8*16*512
    float* agg_im = agg_re + 8 * N_CHUNK * V_DIM;
    float* car_re = agg_im + 8 * N_CHUNK * V_DIM;
    float* car_im = car_re + 8 * N_CHUNK * V_DIM;
    float* y = xs;  // reuse: xs dead after input-projection GEMMs

    // 1. LayerNorm
    ln_kernel<<<M / 8, 256, 0, stream>>>(x, xs);

    // 2. input projection (two GEMMs, gamma epilogue)
    dim3 g1(H_DIM / 64, M / 128);
    gemm_wmma_f32<<<g1, 256, 0, stream>>>(xs, B_re, u_re, M, V_DIM, H_DIM, 1,
                                          gamma_log, nullptr, nullptr);
    gemm_wmma_f32<<<g1, 256, 0, stream>>>(xs, B_im, u_im, M, V_DIM, H_DIM, 1,
                                          gamma_log, nullptr, nullptr);

    // 3. chunked complex scan
    int threadsA = 8 * N_CHUNK * V_DIM;              // 65536
    scan_chunks<<<threadsA / 256, 256, 0, stream>>>(u_re, u_im, nu_log,
                                                    theta_log, agg_re, agg_im);
    scan_carry<<<(8 * V_DIM) / 256, 256, 0, stream>>>(agg_re, agg_im, nu_log,
                                                      theta_log, car_re, car_im);
    scan_apply<<<threadsA / 256, 256, 0, stream>>>(u_re, u_im, nu_log, theta_log,
                                                   car_re, car_im, hcat);

    // 4. output projection: y = hcat @ [C_re ; -C_im]
    pack_b2<<<(2 * V_DIM * H_DIM) / 256, 256, 0, stream>>>(C_re, C_im, B2);
    gemm_wmma_f32<<<g1, 256, 0, stream>>>(hcat, B2, y, M, H_DIM, 2 * V_DIM, 0,
                                          nullptr, nullptr, nullptr);

    // 5. gate: out = y * sigmoid(y @ W^T + b)
    transpose_k<<<dim3(16, 16), 256, 0, stream>>>(W_dense, WT);
    gemm_wmma_f32<<<g1, 256, 0, stream>>>(y, WT, out, M, H_DIM, H_DIM, 2,
                                          nullptr, y, b_dense);
}